// GENPool_18829136625964
// MI455X (gfx1250) — compile-verified
//
#include <hip/hip_runtime.h>
#include <stdint.h>

#define BLOCK 256
#define CHUNK_ROWS 32
#define UNITS_PER_CHUNK (CHUNK_ROWS * 64)           // 2048 x 16-byte units per chunk
#define UNITS_PER_THREAD (UNITS_PER_CHUNK / BLOCK)  // 8 async b128 copies per thread
#define ROWS_PER_THREAD (CHUNK_ROWS / 4)            // 8 rows per thread per chunk

#if defined(__has_builtin)
#if __has_builtin(__builtin_amdgcn_global_load_async_to_lds_b128) && \
    __has_builtin(__builtin_amdgcn_s_wait_asynccnt)
#define USE_ASYNC_LDS 1
#endif
#endif

typedef int v4i __attribute__((ext_vector_type(4)));
typedef __attribute__((address_space(3))) v4i lds_v4i;

// Generic LDS pointers carry the LDS byte offset in their low 32 bits (ISA 10.2
// aperture rules), so the integer round-trip is an exact addrspace cast.
__device__ __forceinline__ lds_v4i* to_lds(void* p) {
  return (lds_v4i*)(uint32_t)(uintptr_t)p;
}

__device__ __forceinline__ void copy16_async(const void* g, void* l) {
#ifdef USE_ASYNC_LDS
  // emits global_load_async_to_lds_b128 (tracked by ASYNCcnt)
  __builtin_amdgcn_global_load_async_to_lds_b128((v4i*)g, to_lds(l), 0, 0);
#else
  *(float4*)l = *(const float4*)g;
#endif
}

template <int N>
__device__ __forceinline__ void wait_async() {
#ifdef USE_ASYNC_LDS
  __builtin_amdgcn_s_wait_asynccnt(N);  // emits s_wait_asynccnt N
#endif
}

// ---------------------------------------------------------------------------
// Pass 1: one streaming sweep over sorted batch[] -> starts[0..B] boundaries.
// starts[b] = first row index i with batch[i] >= b; starts[B] = N.
// Replaces 36 dependent scalar loads per block with 2.
// ---------------------------------------------------------------------------
__global__ __launch_bounds__(BLOCK) void seg_bounds_kernel(
    const int* __restrict__ batch, int* __restrict__ starts, int n, int B) {
  int i = blockIdx.x * BLOCK + threadIdx.x;
  if (i >= n) return;
  int cur  = batch[i];
  int prev = (i == 0) ? -1 : batch[i - 1];
  for (int s = prev + 1; s <= cur; ++s) starts[s] = i;
  if (i == n - 1) {
    for (int s = cur + 1; s <= B; ++s) starts[s] = n;
  }
}

// ---------------------------------------------------------------------------
// Pass 2: one block per segment; single-pass online softmax pooling.
// Rows stream through a 2 x 32 KB LDS double buffer via the async data mover.
// ---------------------------------------------------------------------------
__global__ __launch_bounds__(BLOCK) void genpool_softmax_kernel(
    const float* __restrict__ x, const int* __restrict__ starts,
    const float* __restrict__ p, const float* __restrict__ beta,
    float* __restrict__ out) {
  __shared__ float4 tile[2][UNITS_PER_CHUNK];  // 64 KB double buffer

  const int b   = blockIdx.x;
  const int tid = threadIdx.x;
  const int fq  = tid & 63;  // feature quad: features [4*fq, 4*fq+4)
  const int rl  = tid >> 6;  // row lane 0..3

  const int start = starts[b];
  const int nrows = starts[b + 1] - start;
  const int nch   = (nrows + CHUNK_ROWS - 1) / CHUNK_ROWS;

  const float pv = p[0];
  const float bv = beta[0];
  const float* seg = x + (size_t)start * 256;

  // Online-softmax running state, one slot per owned feature.
  float m[4], d[4], num[4];
#pragma unroll
  for (int k = 0; k < 4; ++k) { m[k] = -3.402823466e38f; d[k] = 0.f; num[k] = 0.f; }

  auto issue = [&](int c, int bi) {
    const int rowBase = c * CHUNK_ROWS;
#pragma unroll
    for (int i = 0; i < UNITS_PER_THREAD; ++i) {
      int u   = i * BLOCK + tid;  // 16-byte unit within chunk
      int row = u >> 6;
      int col = u & 63;
      int gr  = rowBase + row;
      if (gr > nrows - 1) gr = nrows - 1;  // clamp tail: full EXEC, no OOB
      copy16_async(seg + (size_t)gr * 256 + (size_t)col * 4, (void*)&tile[bi][u]);
    }
  };

  int bi = 0;
  if (nch > 0) issue(0, 0);
  for (int c = 0; c < nch; ++c) {
    if (c + 1 < nch) {
      issue(c + 1, bi ^ 1);            // prefetch next chunk into other buffer
      wait_async<UNITS_PER_THREAD>();  // only the next chunk's copies still in flight
    } else {
      wait_async<0>();
    }
    __syncthreads();  // chunk c resident in LDS for all waves

    const float4* buf = &tile[bi][0];
#pragma unroll
    for (int i = 0; i < ROWS_PER_THREAD; ++i) {
      int lr = i * 4 + rl;
      int r  = c * CHUNK_ROWS + lr;
      if (r < nrows) {
        float4 v4 = buf[lr * 64 + fq];  // ds_load_b128, conflict-free
        float v[4] = {v4.x, v4.y, v4.z, v4.w};
#pragma unroll
        for (int k = 0; k < 4; ++k) {
          float a    = pv * v[k];
          float newm = fmaxf(m[k], a);
          float s    = __expf(m[k] - newm);
          float e    = __expf(a - newm);
          d[k]   = d[k] * s + e;
          num[k] = num[k] * s + v[k] * e;
          m[k]   = newm;
        }
      }
    }
    __syncthreads();  // everyone done reading buf before it is re-filled
    bi ^= 1;
  }

  // Merge the 4 row-lane partial softmax states per feature through LDS.
  float* mg = (float*)&tile[0][0];  // 256 threads * 12 floats = 12 KB
#pragma unroll
  for (int k = 0; k < 4; ++k) {
    mg[tid * 12 + k]     = m[k];
    mg[tid * 12 + 4 + k] = d[k];
    mg[tid * 12 + 8 + k] = num[k];
  }
  __syncthreads();

  if (rl == 0) {
    float M[4], D[4], Nm[4];
#pragma unroll
    for (int k = 0; k < 4; ++k) { M[k] = m[k]; D[k] = d[k]; Nm[k] = num[k]; }
    for (int j = 1; j < 4; ++j) {
      const float* o = &mg[(tid + j * 64) * 12];
#pragma unroll
      for (int k = 0; k < 4; ++k) {
        float mo = o[k], d2 = o[4 + k], n2 = o[8 + k];
        float mm = fmaxf(M[k], mo);
        float s1 = __expf(M[k] - mm);
        float s2 = __expf(mo - mm);
        D[k]  = D[k] * s1 + d2 * s2;
        Nm[k] = Nm[k] * s1 + n2 * s2;
        M[k]  = mm;
      }
    }
    const float nn = (float)nrows;
    const float sc = nn / (1.0f + bv * (nn - 1.0f));
    float4 r;
    r.x = (Nm[0] / D[0]) * sc;
    r.y = (Nm[1] / D[1]) * sc;
    r.z = (Nm[2] / D[2]) * sc;
    r.w = (Nm[3] / D[3]) * sc;
    ((float4*)out)[(size_t)b * 64 + fq] = r;
  }
}

extern "C" void kernel_launch(void* const* d_in, const int* in_sizes, int n_in,
                              void* d_out, int out_size, void* d_ws, size_t ws_size,
                              hipStream_t stream) {
  (void)n_in; (void)ws_size;
  const float* x     = (const float*)d_in[0];
  const int*   batch = (const int*)d_in[1];
  const float* p     = (const float*)d_in[2];
  const float* beta  = (const float*)d_in[3];
  float* out = (float*)d_out;
  int*   starts = (int*)d_ws;  // (B+1) ints of scratch

  const int n = in_sizes[1];        // N rows
  const int B = out_size / 256;     // segments (512)

  hipLaunchKernelGGL(seg_bounds_kernel, dim3((n + BLOCK - 1) / BLOCK), dim3(BLOCK),
                     0, stream, batch, starts, n, B);
  hipLaunchKernelGGL(genpool_softmax_kernel, dim3(B), dim3(BLOCK), 0, stream,
                     x, starts, p, beta, out);
}